// SimSiamAdvMMD_28913719837153
// MI455X (gfx1250) — compile-verified
//
#include <hip/hip_runtime.h>
#include <hip/hip_bf16.h>

// ---------------------------------------------------------------------------
// SimSiam adversarial KMM loss, restructured for MI455X (gfx1250, wave32).
//   n = 256 samples, d = 128 features.
// Key algebraic collapse: the 256 per-sample kernel matrices are identical,
// and the per-sample RHS batch is a single GEMM  B_s = E_s * W_s.
// All GEMMs use V_WMMA_F32_16X16X4_F32 (f32 matrix core, exact).
// ---------------------------------------------------------------------------

typedef float v2f __attribute__((ext_vector_type(2)));
typedef float v8f __attribute__((ext_vector_type(8)));

static constexpr int   NN  = 256;
static constexpr int   DD  = 128;
static constexpr size_t T  = (size_t)NN * NN;   // 65536 floats per 256x256 tile
static constexpr float LAM = 1e-3f;
static constexpr float EPS = 1e-3f;

__device__ __constant__ float d_i2s2[4] = {0.5f, 0.005f, 5.0e-5f, 5.0e-7f};

// ---------------------------------------------------------------------------
// 0) Gather + build concatenated embeddings.
//    Xc[i] = [z1_i | z2_i]        (numerator x_de)
//    Yc[i] = [z1s_i | z2s_i]      (numerator x_nu)
//    z2s   = z2[perm2]
// grid=256 (row), block=128
__global__ void k_prep(const float* __restrict__ z1, const float* __restrict__ z2,
                       const int* __restrict__ p1, const int* __restrict__ p2,
                       float* __restrict__ Xc, float* __restrict__ Yc,
                       float* __restrict__ z2s) {
  int i = blockIdx.x, t = threadIdx.x;
  int a = p1[i], b = p2[i];
  Xc[i * NN + t]      = z1[i * DD + t];
  Xc[i * NN + DD + t] = z2[i * DD + t];
  Yc[i * NN + t]      = z1[a * DD + t];
  Yc[i * NN + DD + t] = z2[b * DD + t];
  z2s[i * DD + t]     = z2[b * DD + t];
}

// 1) Row squared norms. grid=1, block=256
__global__ void k_norms(const float* __restrict__ Xc, const float* __restrict__ Yc,
                        const float* __restrict__ z1, const float* __restrict__ z2,
                        const float* __restrict__ z2s, float* __restrict__ sm) {
  int i = threadIdx.x;
  float sx = 0.f, sy = 0.f;
  for (int k = 0; k < NN; ++k) { float a = Xc[i*NN+k], b = Yc[i*NN+k]; sx += a*a; sy += b*b; }
  float s1 = 0.f, s2 = 0.f, ss = 0.f;
  for (int k = 0; k < DD; ++k) {
    float a = z1[i*DD+k], b = z2[i*DD+k], c = z2s[i*DD+k];
    s1 += a*a; s2 += b*b; ss += c*c;
  }
  sm[i] = sx; sm[256+i] = sy; sm[512+i] = s1; sm[768+i] = s2; sm[1024+i] = ss;
}

// ---------------------------------------------------------------------------
// 2) WMMA f32 GEMM.
//    TRANSB=true : C[M,N] = A[M,K] * B[N,K]^T   (Gram matrices)
//    TRANSB=false: C[M,N] = A[M,K] * B[K,N]     (E * W batch)
// One wave (32 threads) per 16x16 C tile, K stepped by 4.
// A 16x4 f32 layout: lanes 0-15 -> K={0,1}, lanes 16-31 -> K={2,3}, M = lane&15.
// grid=(N/16, M/16, batch), block=32
template <bool TRANSB>
__global__ void k_gemm_wmma(const float* __restrict__ A, const float* __restrict__ B,
                            float* __restrict__ C, int M, int Nc, int K,
                            size_t sA, size_t sB, size_t sC) {
  A += blockIdx.z * sA; B += blockIdx.z * sB; C += blockIdx.z * sC;
  int tn = blockIdx.x, tm = blockIdx.y;
  int lane = threadIdx.x;
  int ml   = lane & 15;
  int ksel = (lane >> 4) << 1;           // 0 or 2
  v8f acc = {};
  for (int k0 = 0; k0 < K; k0 += 4) {
    v2f av, bv;
    const float* ar = A + (size_t)(tm * 16 + ml) * K + k0 + ksel;
    av.x = ar[0]; av.y = ar[1];
    if (TRANSB) {
      const float* br = B + (size_t)(tn * 16 + ml) * K + k0 + ksel;
      bv.x = br[0]; bv.y = br[1];
    } else {
      bv.x = B[(size_t)(k0 + ksel)     * Nc + tn * 16 + ml];
      bv.y = B[(size_t)(k0 + ksel + 1) * Nc + tn * 16 + ml];
    }
    acc = __builtin_amdgcn_wmma_f32_16x16x4_f32(false, av, false, bv,
                                                (short)0, acc, false, false);
  }
  int rbase = (lane < 16) ? 0 : 8;
  int col   = tn * 16 + ml;
  for (int r = 0; r < 8; ++r)
    C[(size_t)(tm * 16 + rbase + r) * Nc + col] = acc[r];
}

// ---------------------------------------------------------------------------
// 3) Numerator kernel matrices: A_num_s = exp(-max(0,dsq_XX)*i2s2) + lam*I
// grid=256 (i), block=256 (j)
__global__ void k_build_anum(const float* __restrict__ Gxx, const float* __restrict__ sm,
                             float* __restrict__ Anum) {
  int i = blockIdx.x, j = threadIdx.x;
  float d = fmaxf(sm[i] + sm[j] - 2.f * Gxx[i * NN + j], 0.f);
  float dg = (i == j) ? LAM : 0.f;
#pragma unroll
  for (int s = 0; s < 4; ++s)
    Anum[s * T + i * NN + j] = __expf(-d * d_i2s2[s]) + dg;
}

// 4) Numerator RHS: b_s[i] = sum_j exp(-max(0,dsq_XY[i,j])*i2s2)
// grid=4 (sigma), block=256 (i)
__global__ void k_bnum(const float* __restrict__ Gxy, const float* __restrict__ sm,
                       float* __restrict__ bnum) {
  int s = blockIdx.x, i = threadIdx.x;
  float i2 = d_i2s2[s], nXi = sm[i], acc = 0.f;
  for (int j = 0; j < NN; ++j) {
    float d = fmaxf(nXi + sm[256 + j] - 2.f * Gxy[i * NN + j], 0.f);
    acc += __expf(-d * i2);
  }
  bnum[s * NN + i] = acc;
}

// 5) Denominator: E_s[j,k] = exp(-e*i2s2), A_den_s = exp(-max(0,dsq22))+lam*I
//    e[j,k] = |z2s_j|^2 + 2|z2_k|^2 - 2 z2s_j.z2_k
// grid=256 (j), block=256 (k)
__global__ void k_build_den(const float* __restrict__ G22, const float* __restrict__ G2z,
                            const float* __restrict__ sm,
                            float* __restrict__ E, float* __restrict__ Aden) {
  int j = blockIdx.x, k = threadIdx.x;
  float n2s_j = sm[1024 + j];
  float e  = n2s_j + 2.f * sm[768 + k] - 2.f * G2z[j * NN + k];
  float dd = fmaxf(n2s_j + sm[1024 + k] - 2.f * G22[j * NN + k], 0.f);
  float dg = (j == k) ? LAM : 0.f;
#pragma unroll
  for (int s = 0; s < 4; ++s) {
    E[s * T + j * NN + k]    = __expf(-e  * d_i2s2[s]);
    Aden[s * T + j * NN + k] = __expf(-dd * d_i2s2[s]) + dg;
  }
}

// 6) W_s[k,i] = exp(-(|z1_i|^2 - 2 z1_i.z2_k)*i2s2).  grid=256 (k), block=256 (i)
__global__ void k_build_w(const float* __restrict__ G1z, const float* __restrict__ sm,
                          float* __restrict__ W) {
  int k = blockIdx.x, i = threadIdx.x;
  float c = sm[512 + i] - 2.f * G1z[i * NN + k];
#pragma unroll
  for (int s = 0; s < 4; ++s)
    W[s * T + k * NN + i] = __expf(-c * d_i2s2[s]);
}

// ---------------------------------------------------------------------------
// 7) In-place Cholesky (lower) of one 256x256 SPD matrix per block.
// Matrices live in L2; one 256-thread block each. grid=8, block=256
__global__ void k_chol(float* __restrict__ Aall) {
  float* A = Aall + (size_t)blockIdx.x * T;
  int t = threadIdx.x;
  for (int j = 0; j < NN; ++j) {
    if (t == 0) A[j * NN + j] = sqrtf(fmaxf(A[j * NN + j], 1e-30f));
    __syncthreads();
    float dj = A[j * NN + j];
    for (int r = j + 1 + t; r < NN; r += 256) A[r * NN + j] /= dj;
    __syncthreads();
    int nt = NN - 1 - j;
    for (int idx = t; idx < nt * nt; idx += 256) {
      int r = j + 1 + idx / nt, k = j + 1 + idx % nt;
      if (k <= r) A[r * NN + k] -= A[r * NN + j] * A[k * NN + j];
    }
    __syncthreads();
  }
}

// 8) Triangular solves, in place over B: X = (L L^T)^{-1} B.
// One thread per RHS column; fully independent (no syncs), coalesced reads.
// grid=4 (sigma), block=ncols
__global__ void k_trisolve(const float* __restrict__ Lall, size_t lstride,
                           float* __restrict__ Ball, size_t bstride,
                           int ldb, int ncols) {
  const float* L = Lall + (size_t)blockIdx.x * lstride;
  float* B = Ball + (size_t)blockIdx.x * bstride;
  int c = threadIdx.x;
  if (c >= ncols) return;
  for (int j = 0; j < NN; ++j) {                 // forward:  L y = b
    float acc = B[j * ldb + c];
    for (int k = 0; k < j; ++k) acc -= L[j * NN + k] * B[k * ldb + c];
    B[j * ldb + c] = acc / L[j * NN + j];
  }
  for (int j = NN - 1; j >= 0; --j) {            // backward: L^T x = y
    float acc = B[j * ldb + c];
    for (int k = j + 1; k < NN; ++k) acc -= L[k * NN + j] * B[k * ldb + c];
    B[j * ldb + c] = acc / L[j * NN + j];
  }
}

// ---------------------------------------------------------------------------
// 9) Final loss. grid=1, block=256
//    loss = sum_i [ log(relu(avg_s xnum_s[i]) + eps) + log(sum_j relu(avg_s X_s[j,i]) + eps) ]
__global__ void k_final(const float* __restrict__ X,     // 4*T, [s][j*256+i]
                        const float* __restrict__ xnum,  // 4*256, [s*256+i]
                        float* __restrict__ out) {
  __shared__ float red[256];
  int i = threadIdx.x;
  float dsum = 0.f;
  for (int j = 0; j < NN; ++j) {
    float r = 0.25f * (X[j * NN + i] + X[T + j * NN + i] +
                       X[2 * T + j * NN + i] + X[3 * T + j * NN + i]);
    dsum += fmaxf(r, 0.f) + EPS;
  }
  float rn = 0.25f * (xnum[i] + xnum[256 + i] + xnum[512 + i] + xnum[768 + i]);
  red[i] = __logf(dsum) + __logf(fmaxf(rn, 0.f) + EPS);
  __syncthreads();
  for (int off = 128; off; off >>= 1) {
    if (i < off) red[i] += red[i + off];
    __syncthreads();
  }
  if (i == 0) out[0] = red[0];
}

// ---------------------------------------------------------------------------
extern "C" void kernel_launch(void* const* d_in, const int* in_sizes, int n_in,
                              void* d_out, int out_size, void* d_ws, size_t ws_size,
                              hipStream_t stream) {
  const float* z1 = (const float*)d_in[0];
  const float* z2 = (const float*)d_in[1];
  const int* p1   = (const int*)d_in[2];
  const int* p2   = (const int*)d_in[3];
  float* out = (float*)d_out;
  float* w = (float*)d_ws;

  // workspace layout (floats)
  float* Xc   = w + 0 * T;
  float* Yc   = w + 1 * T;
  float* z2s  = w + 2 * T;          // 256x128 (half tile used)
  float* Gxx  = w + 3 * T;
  float* Gxy  = w + 4 * T;
  float* G22  = w + 5 * T;
  float* G2z  = w + 6 * T;
  float* G1z  = w + 7 * T;
  float* E    = w + 8 * T;          // 4*T
  float* W    = w + 12 * T;         // 4*T
  float* Aall = w + 16 * T;         // 8*T : [0..3]=A_num, [4..7]=A_den
  float* B    = w + 24 * T;         // 4*T : RHS batch -> solved ratios in place
  float* sm   = w + 28 * T;         // norms (5*256) ...
  float* bnum = sm + 1280;          // 4*256 numerator RHS -> solved in place

  dim3 g16(NN / 16, NN / 16, 1);

  k_prep<<<NN, DD, 0, stream>>>(z1, z2, p1, p2, Xc, Yc, z2s);
  k_norms<<<1, NN, 0, stream>>>(Xc, Yc, z1, z2, z2s, sm);

  // Gram matrices (WMMA f32): C = A * B^T
  k_gemm_wmma<true><<<g16, 32, 0, stream>>>(Xc,  Xc,  Gxx, NN, NN, NN, 0, 0, 0);
  k_gemm_wmma<true><<<g16, 32, 0, stream>>>(Xc,  Yc,  Gxy, NN, NN, NN, 0, 0, 0);
  k_gemm_wmma<true><<<g16, 32, 0, stream>>>(z2s, z2s, G22, NN, NN, DD, 0, 0, 0);
  k_gemm_wmma<true><<<g16, 32, 0, stream>>>(z2s, z2,  G2z, NN, NN, DD, 0, 0, 0);
  k_gemm_wmma<true><<<g16, 32, 0, stream>>>(z1,  z2,  G1z, NN, NN, DD, 0, 0, 0);

  // kernel matrices + RHS
  k_build_anum<<<NN, NN, 0, stream>>>(Gxx, sm, Aall);            // A_num (4)
  k_bnum<<<4, NN, 0, stream>>>(Gxy, sm, bnum);                   // b_num (4x256)
  k_build_den<<<NN, NN, 0, stream>>>(G22, G2z, sm, E, Aall + 4 * T);
  k_build_w<<<NN, NN, 0, stream>>>(G1z, sm, W);

  // batched RHS: B_s = E_s * W_s (WMMA f32, batch over sigma)
  dim3 gb(NN / 16, NN / 16, 4);
  k_gemm_wmma<false><<<gb, 32, 0, stream>>>(E, W, B, NN, NN, NN, T, T, T);

  // factor all 8 SPD matrices, then solve
  k_chol<<<8, NN, 0, stream>>>(Aall);
  k_trisolve<<<4, NN, 0, stream>>>(Aall + 4 * T, T, B, T, NN, NN);   // denominator, 256 RHS
  k_trisolve<<<4, 1, 0, stream>>>(Aall, T, bnum, (size_t)NN, 1, 1);  // numerator, 1 RHS

  k_final<<<1, NN, 0, stream>>>(B, bnum, out);
  (void)in_sizes; (void)n_in; (void)out_size; (void)ws_size;
}